// GNN_12103217840680
// MI455X (gfx1250) — compile-verified
//
#include <hip/hip_runtime.h>
#include <cstdint>
#include <cstddef>

#define NNODES 50000
#define DIM    128
#define NEDGES 600000
#define LDS_STR 132   // 128 + 4-word pad: (4*row + k) mod 64 banks -> conflict-free

typedef __attribute__((ext_vector_type(2))) float v2f;
typedef __attribute__((ext_vector_type(8))) float v8f;

__global__ void zero_f32(float* __restrict__ p, int n) {
    int i = blockIdx.x * blockDim.x + threadIdx.x;
    if (i < n) p[i] = 0.0f;
}

__global__ void degree_kernel(const int* __restrict__ dst,
                              float* __restrict__ cnt, int E) {
    int e = blockIdx.x * blockDim.x + threadIdx.x;
    if (e < E) atomicAdd(&cnt[dst[e]], 1.0f);
}

// 32 threads per edge, each handles a float4 chunk of the 128-wide feature row.
__global__ void scatter_add_kernel(const float* __restrict__ h,
                                   const int* __restrict__ src,
                                   const int* __restrict__ dst,
                                   float* __restrict__ agg, int E) {
    int t = blockIdx.x * blockDim.x + threadIdx.x;
    int e = t >> 5;
    if (e >= E) return;
    int c = (t & 31) << 2;
    const float4 v = *reinterpret_cast<const float4*>(h + (size_t)src[e] * DIM + c);
    float* p = agg + (size_t)dst[e] * DIM + c;
    atomicAdd(p + 0, v.x);
    atomicAdd(p + 1, v.y);
    atomicAdd(p + 2, v.z);
    atomicAdd(p + 3, v.w);
}

// Fused SAGE layer: out = (agg * 1/max(cnt,1)) @ Wl + h @ Wr + b  [, ReLU]
// One block = 16 output rows; 8 waves = 8 column tiles of 16.
// A-slabs (scaled agg + h, 16x128 each) staged in LDS, padded stride.
__global__ __launch_bounds__(256)
void sage_gemm_kernel(const float* __restrict__ agg,
                      const float* __restrict__ h,
                      const float* __restrict__ cnt,
                      const float* __restrict__ Wl,
                      const float* __restrict__ Wr,
                      const float* __restrict__ bias,
                      float* __restrict__ out,
                      int relu) {
    __shared__ float sA[16 * LDS_STR];   // degree-normalized agg rows
    __shared__ float sH[16 * LDS_STR];   // h rows

    const int rowbase = blockIdx.x * 16;            // 3125 blocks * 16 = 50000 exact
    const int tid = threadIdx.x;

    // Cooperative stage of the 16-row A slabs (fuses the mean-division).
    for (int i = tid; i < 16 * DIM; i += 256) {
        int r = i >> 7;          // row within tile
        int c = i & (DIM - 1);   // column
        int row = rowbase + r;
        float inv = 1.0f / fmaxf(cnt[row], 1.0f);
        sA[r * LDS_STR + c] = agg[(size_t)row * DIM + c] * inv;
        sH[r * LDS_STR + c] = h[(size_t)row * DIM + c];
    }
    __syncthreads();

    const int wave    = tid >> 5;          // 0..7 -> column tile
    const int lane    = tid & 31;
    const int colbase = wave * 16;
    const int mrow    = lane & 15;         // M row (A) / N col (B,C,D)
    const int khalf   = (lane >> 4) * 2;   // lane group K offset: 0 or 2

    __builtin_prefetch(&Wl[colbase], 0, 3);
    __builtin_prefetch(&Wr[colbase], 0, 3);

    v8f acc = {};

    // acc += agg_tile @ Wl   (exact fp32 WMMA, K in steps of 4)
    #pragma unroll
    for (int k = 0; k < DIM; k += 4) {
        v2f a = *reinterpret_cast<const v2f*>(&sA[mrow * LDS_STR + k + khalf]);
        v2f b;
        b.x = Wl[(size_t)(k + khalf + 0) * DIM + colbase + mrow];
        b.y = Wl[(size_t)(k + khalf + 1) * DIM + colbase + mrow];
        acc = __builtin_amdgcn_wmma_f32_16x16x4_f32(
            false, a, false, b, (short)0, acc, false, false);
    }
    // acc += h_tile @ Wr
    #pragma unroll
    for (int k = 0; k < DIM; k += 4) {
        v2f a = *reinterpret_cast<const v2f*>(&sH[mrow * LDS_STR + k + khalf]);
        v2f b;
        b.x = Wr[(size_t)(k + khalf + 0) * DIM + colbase + mrow];
        b.y = Wr[(size_t)(k + khalf + 1) * DIM + colbase + mrow];
        acc = __builtin_amdgcn_wmma_f32_16x16x4_f32(
            false, a, false, b, (short)0, acc, false, false);
    }

    // C/D layout: VGPR i -> M = i + 8*(lane>=16); N = lane&15 (+colbase)
    const int rshift = (lane >> 4) * 8;
    const int col    = colbase + mrow;
    const float bv   = bias[col];
    #pragma unroll
    for (int i = 0; i < 8; ++i) {
        int row = rowbase + i + rshift;
        float v = acc[i] + bv;
        if (relu) v = fmaxf(v, 0.0f);
        out[(size_t)row * DIM + col] = v;
    }
}

extern "C" void kernel_launch(void* const* d_in, const int* in_sizes, int n_in,
                              void* d_out, int out_size, void* d_ws, size_t ws_size,
                              hipStream_t stream) {
    const float* x    = (const float*)d_in[0];
    const int*   eidx = (const int*)d_in[1];          // [2, E] flat: row0=src, row1=dst
    const float* Wl1  = (const float*)d_in[2];
    const float* Wr1  = (const float*)d_in[3];
    const float* b1   = (const float*)d_in[4];
    const float* Wl2  = (const float*)d_in[5];
    const float* Wr2  = (const float*)d_in[6];
    const float* b2   = (const float*)d_in[7];
    float*       outp = (float*)d_out;

    const int* src = eidx;
    const int* dst = eidx + NEDGES;

    // Workspace layout (floats): agg [N*D] | h1 [N*D] | cnt [N]
    float* agg = (float*)d_ws;
    float* h1  = agg + (size_t)NNODES * DIM;
    float* cnt = h1 + (size_t)NNODES * DIM;

    const int ND = NNODES * DIM;
    dim3 blk(256);

    // ---- Layer 1 ----
    zero_f32<<<dim3((ND + 255) / 256), blk, 0, stream>>>(agg, ND);
    zero_f32<<<dim3((NNODES + 255) / 256), blk, 0, stream>>>(cnt, NNODES);
    degree_kernel<<<dim3((NEDGES + 255) / 256), blk, 0, stream>>>(dst, cnt, NEDGES);
    scatter_add_kernel<<<dim3((NEDGES * 32 + 255) / 256), blk, 0, stream>>>(
        x, src, dst, agg, NEDGES);
    sage_gemm_kernel<<<dim3(NNODES / 16), blk, 0, stream>>>(
        agg, x, cnt, Wl1, Wr1, b1, h1, /*relu=*/1);

    // ---- Layer 2 ----
    zero_f32<<<dim3((ND + 255) / 256), blk, 0, stream>>>(agg, ND);
    scatter_add_kernel<<<dim3((NEDGES * 32 + 255) / 256), blk, 0, stream>>>(
        h1, src, dst, agg, NEDGES);
    sage_gemm_kernel<<<dim3(NNODES / 16), blk, 0, stream>>>(
        agg, h1, cnt, Wl2, Wr2, b2, outp, /*relu=*/0);
}